// CoAttn_81174881895059
// MI455X (gfx1250) — compile-verified
//
#include <hip/hip_runtime.h>
#include <hip/hip_bf16.h>

typedef __attribute__((ext_vector_type(2))) float v2f;
typedef __attribute__((ext_vector_type(8))) float v8f;

#define NF   512
#define SP   49
#define LDQ  52      // padded K stride for Qa/Qb rows
#define LDA  64      // padded softmax tile stride
#define NTHREADS 256

__device__ __forceinline__ float blockReduceMax(float v, float* sRed, int tid) {
    sRed[tid] = v;
    __syncthreads();
    for (int s = NTHREADS / 2; s > 0; s >>= 1) {
        if (tid < s) sRed[tid] = fmaxf(sRed[tid], sRed[tid + s]);
        __syncthreads();
    }
    float r = sRed[0];
    __syncthreads();
    return r;
}

__device__ __forceinline__ float blockReduceSum(float v, float* sRed, int tid) {
    sRed[tid] = v;
    __syncthreads();
    for (int s = NTHREADS / 2; s > 0; s >>= 1) {
        if (tid < s) sRed[tid] += sRed[tid + s];
        __syncthreads();
    }
    float r = sRed[0];
    __syncthreads();
    return r;
}

__global__ __launch_bounds__(NTHREADS) void coattn_kernel(
        const float* __restrict__ x,   // (B, 2, NF, 7, 7)
        const float* __restrict__ W,   // (49, 49)
        float* __restrict__ out) {     // (B, 2, NF, 7, 7)
    extern __shared__ float smem[];
    float* sQa  = smem;                    // NF*LDQ
    float* sQb  = sQa + NF * LDQ;          // NF*LDQ
    float* sA   = sQb + NF * LDQ;          // LDA*LDA, A (zero padded)
    float* sAT  = sA + LDA * LDA;          // LDA*LDA, A^T (zero padded)
    float* sq   = sAT + LDA * LDA;         // qa[64] | qb[64]
    float* sRed = sq + 128;                // NTHREADS

    const int tid = threadIdx.x;
    const int b   = blockIdx.x;
    const float* xA = x + (size_t)b * 2u * NF * SP;
    const float* xB = xA + NF * SP;

    // ---- init padded regions ----
    for (int i = tid; i < LDA * LDA; i += NTHREADS) { sA[i] = 0.0f; sAT[i] = 0.0f; }
    if (tid < 128) sq[tid] = 0.0f;

    // ---- stage Qa/Qb into LDS (coalesced float4 global reads) ----
    const int NV = NF * SP / 4;            // 6272 float4 per matrix
    for (int i = tid; i < NV; i += NTHREADS) {
        float4 va = ((const float4*)xA)[i];
        float4 vb = ((const float4*)xB)[i];
        float av[4] = {va.x, va.y, va.z, va.w};
        float bv[4] = {vb.x, vb.y, vb.z, vb.w};
        int base = i * 4;
#pragma unroll
        for (int e = 0; e < 4; ++e) {
            int flat = base + e;
            int r = flat / SP;
            int c = flat - r * SP;
            sQa[r * LDQ + c] = av[e];
            sQb[r * LDQ + c] = bv[e];
        }
    }
    for (int r = tid; r < NF; r += NTHREADS) {
        sQa[r * LDQ + 49] = 0.0f; sQa[r * LDQ + 50] = 0.0f; sQa[r * LDQ + 51] = 0.0f;
        sQb[r * LDQ + 49] = 0.0f; sQb[r * LDQ + 50] = 0.0f; sQb[r * LDQ + 51] = 0.0f;
    }
    __syncthreads();

    // ---- qa/qb = column-wise abs sums (2 mats * 49 cols * 8 row-groups) ----
    for (int t = tid; t < 2 * SP * 8; t += NTHREADS) {
        int mat = (t >= SP * 8) ? 1 : 0;
        int rem = t - mat * SP * 8;
        int col = rem % SP;
        int grp = rem / SP;                  // 0..7, 64 rows each
        const float* Q = mat ? sQb : sQa;
        float s = 0.0f;
        int r0 = grp * 64;
        for (int r = 0; r < 64; ++r) s += fabsf(Q[(r0 + r) * LDQ + col]);
        atomicAdd(&sq[mat * 64 + col], s);   // ds_add_f32
    }
    __syncthreads();

    // ---- L = qa[i]*W[j,i]*qb[j]; softmax over all 2401 entries ----
    float lmax = -3.402823466e38f;
    for (int idx = tid; idx < SP * SP; idx += NTHREADS) {
        int j = idx / SP;
        int i = idx - j * SP;
        float v = sq[i] * W[j * SP + i] * sq[64 + j];
        sA[j * LDA + i] = v;
        lmax = fmaxf(lmax, v);
    }
    float mx = blockReduceMax(lmax, sRed, tid);

    float lsum = 0.0f;
    for (int idx = tid; idx < SP * SP; idx += NTHREADS) {
        int j = idx / SP;
        int i = idx - j * SP;
        float e = __expf(sA[j * LDA + i] - mx);
        sA[j * LDA + i] = e;
        lsum += e;
    }
    float inv = 1.0f / blockReduceSum(lsum, sRed, tid);
    for (int idx = tid; idx < SP * SP; idx += NTHREADS) {
        int j = idx / SP;
        int i = idx - j * SP;
        float e = sA[j * LDA + i] * inv;
        sA [j * LDA + i] = e;                // A   (B for Za)
        sAT[i * LDA + j] = e;                // A^T (B for Zb)
    }
    __syncthreads();

    // ---- GEMMs: Za = Qa @ A ; Zb = Qb @ A^T via V_WMMA_F32_16X16X4_F32 ----
    // 32 M-tiles * 4 N-tiles * 2 outputs = 256 tiles across 8 waves
    const int wave = tid >> 5;
    const int lane = tid & 31;
    const int half = lane >> 4;    // 0: K+{0,1} ; 1: K+{2,3}
    const int nl   = lane & 15;    // M-row within A-frag / N-col within B-frag

    for (int tile = wave; tile < 256; tile += 8) {
        int p  = tile & 1;                    // 0 -> Za, 1 -> Zb
        int n0 = ((tile >> 1) & 3) << 4;      // N tile origin (0..48)
        int m0 = (tile >> 3) << 4;            // M tile origin (0..496)
        const float* Q  = p ? sQb : sQa;      // A-matrix source
        const float* Bm = p ? sAT : sA;       // B-matrix source (row-major K x N)

        const int arow = (m0 + nl) * LDQ;     // A-frag row base
        const int bcol = n0 + nl;             // B-frag column

        v8f acc = {};
#pragma unroll
        for (int k = 0; k < LDQ; k += 4) {    // 13 k-steps, zero-padded tail
            int ka = k + 2 * half;
            v2f a, bf;
            a[0]  = Q[arow + ka];
            a[1]  = Q[arow + ka + 1];
            bf[0] = Bm[ka * LDA + bcol];
            bf[1] = Bm[(ka + 1) * LDA + bcol];
            acc = __builtin_amdgcn_wmma_f32_16x16x4_f32(
                false, a, false, bf, (short)0, acc, false, false);
        }

        // C/D layout: VGPR r -> M = m0 + r + 8*half, N = n0 + (lane&15)
        if (bcol < SP) {
            size_t obase = ((size_t)(b * 2 + p) * NF) * SP;
#pragma unroll
            for (int r = 0; r < 8; ++r) {
                int row = m0 + r + half * 8;
                out[obase + (size_t)row * SP + bcol] = acc[r];
            }
        }
    }
}

extern "C" void kernel_launch(void* const* d_in, const int* in_sizes, int n_in,
                              void* d_out, int out_size, void* d_ws, size_t ws_size,
                              hipStream_t stream) {
    const float* x = (const float*)d_in[0];
    const float* W = (const float*)d_in[1];
    float* out = (float*)d_out;

    int B = in_sizes[0] / (2 * NF * SP);   // 2048
    size_t smemBytes = (size_t)(2 * NF * LDQ + 2 * LDA * LDA + 128 + NTHREADS) * sizeof(float);

    coattn_kernel<<<B, NTHREADS, smemBytes, stream>>>(x, W, out);
}